// Sparsemax_35287451304533
// MI455X (gfx1250) — compile-verified
//
#include <hip/hip_runtime.h>

#define ROW_N 1024
#define WAVES_PER_BLOCK 8
#define BLOCK_THREADS (WAVES_PER_BLOCK * 32)
#define BISECT_ITERS 24

// ---- gfx1250 async global->LDS path (guarded; fallback = direct loads) ----
#if __has_builtin(__builtin_amdgcn_global_load_async_to_lds_b128)
#define HAVE_ASYNC_LDS 1
#else
#define HAVE_ASYNC_LDS 0
#endif

#if __has_builtin(__builtin_amdgcn_s_wait_asynccnt)
#define WAIT_ASYNC(n) __builtin_amdgcn_s_wait_asynccnt(n)
#else
#define WAIT_ASYNC(n) asm volatile("s_wait_asynccnt %0" ::"i"(n) : "memory")
#endif

typedef int v4i __attribute__((ext_vector_type(4)));
typedef __attribute__((address_space(1))) v4i* global_v4i_ptr;
typedef __attribute__((address_space(3))) v4i* lds_v4i_ptr;

__device__ __forceinline__ void async_copy_row(const float* __restrict__ gsrc,
                                               float* lds_dst, int lane) {
#if HAVE_ASYNC_LDS
#pragma unroll
  for (int c = 0; c < 8; ++c) {
    // each lane moves 16B; one issue moves 512B for the wave; 8 issues = 4KB row
    __builtin_amdgcn_global_load_async_to_lds_b128(
        (global_v4i_ptr)(gsrc + c * 128 + lane * 4),
        (lds_v4i_ptr)(lds_dst + c * 128 + lane * 4),
        0, 0);
  }
#else
  (void)gsrc; (void)lds_dst; (void)lane;
#endif
}

__device__ __forceinline__ float wave_reduce_add(float v) {
#pragma unroll
  for (int k = 16; k; k >>= 1) v += __shfl_xor(v, k, 32);
  return v;  // butterfly: every lane gets the total (uniform decisions)
}

__device__ __forceinline__ float wave_reduce_max(float v) {
#pragma unroll
  for (int k = 16; k; k >>= 1) v = fmaxf(v, __shfl_xor(v, k, 32));
  return v;
}

__global__ __launch_bounds__(BLOCK_THREADS)
void sparsemax_kernel(const float* __restrict__ x, float* __restrict__ out, int rows) {
  __shared__ float lds[2][WAVES_PER_BLOCK][ROW_N];  // 64 KB: double-buffered row stage

  const int lane = threadIdx.x & 31;
  const int w    = threadIdx.x >> 5;
  const unsigned lt_mask = (1u << lane) - 1u;
  const long total_waves = (long)gridDim.x * WAVES_PER_BLOCK;
  const long row0 = (long)blockIdx.x * WAVES_PER_BLOCK + w;

#if HAVE_ASYNC_LDS
  if (row0 < rows) async_copy_row(x + row0 * ROW_N, &lds[0][w][0], lane);
#endif

  int buf = 0;
  for (long row = row0; row < rows; row += total_waves) {
    float* rowbuf = &lds[buf][w][0];

#if HAVE_ASYNC_LDS
    // software pipeline: kick off next row's DMA, then wait for current row
    const long nxt = row + total_waves;
    if (nxt < (long)rows) {
      async_copy_row(x + nxt * ROW_N, &lds[buf ^ 1][w][0], lane);
      WAIT_ASYNC(8);   // 8 outstanding = the next row; current row complete
    } else {
      WAIT_ASYNC(0);
    }
#endif

    // ---- load 32 elements/lane into registers, track max ----
    float r[32];
    float m = -3.402823466e38f;
#if HAVE_ASYNC_LDS
#pragma unroll
    for (int c = 0; c < 8; ++c) {
      float4 t = *(const float4*)(rowbuf + c * 128 + lane * 4);  // ds_load_b128
      r[4 * c + 0] = t.x; r[4 * c + 1] = t.y; r[4 * c + 2] = t.z; r[4 * c + 3] = t.w;
    }
#else
    const float4* gp = (const float4*)(x + row * ROW_N);
#pragma unroll
    for (int c = 0; c < 8; ++c) {
      float4 t = gp[c * 32 + lane];  // global_load_b128, coalesced
      r[4 * c + 0] = t.x; r[4 * c + 1] = t.y; r[4 * c + 2] = t.z; r[4 * c + 3] = t.w;
    }
#endif
#pragma unroll
    for (int i = 0; i < 32; ++i) m = fmaxf(m, r[i]);
    m = wave_reduce_max(m);

    // ---- compact active set {x > max-1} into LDS (reuse row buffer) ----
    // tau >= max-1 always, so elements <= max-1 contribute 0 to every f(tau).
    const float thr = m - 1.0f;
    float* act = rowbuf;  // registers hold the row now; LDS slot is free
    int cnt = 0;
#pragma unroll
    for (int i = 0; i < 32; ++i) {
      const float v = r[i];
      const bool p = v > thr;
      const unsigned msk = (unsigned)__ballot(p);  // wave32: low 32 bits
      if (p) act[cnt + __popc(msk & lt_mask)] = v;
      cnt += __popc(msk);
    }

    // ---- bisection on tau over [max-1, max], then exact snap ----
    float lo = thr, hi = m, tau;
    if (cnt <= 32) {  // common case for Gaussian rows (~14 actives)
      const float a = (lane < cnt) ? act[lane] : thr;  // pad contributes 0
#pragma unroll 1
      for (int it = 0; it < BISECT_ITERS; ++it) {
        const float mid = 0.5f * (lo + hi);
        const float s = wave_reduce_add(fmaxf(a - mid, 0.0f));
        if (s >= 1.0f) lo = mid; else hi = mid;
      }
      const bool p = a > lo;
      const float sv = wave_reduce_add(p ? a : 0.0f);
      const float kc = wave_reduce_add(p ? 1.0f : 0.0f);
      tau = (sv - 1.0f) / kc;  // kc >= 1: the row max is always active
    } else {          // rare: nearly-flat row; iterate compacted LDS chunks
#pragma unroll 1
      for (int it = 0; it < BISECT_ITERS; ++it) {
        const float mid = 0.5f * (lo + hi);
        float s = 0.0f;
        for (int i = lane; i < cnt; i += 32) s += fmaxf(act[i] - mid, 0.0f);
        s = wave_reduce_add(s);
        if (s >= 1.0f) lo = mid; else hi = mid;
      }
      float sv = 0.0f, kc = 0.0f;
      for (int i = lane; i < cnt; i += 32) {
        const float v = act[i];
        if (v > lo) { sv += v; kc += 1.0f; }
      }
      sv = wave_reduce_add(sv);
      kc = wave_reduce_add(kc);
      tau = (sv - 1.0f) / kc;
    }

    // ---- streaming output: max(x - tau, 0) ----
    float4* op = (float4*)(out + row * ROW_N);
#pragma unroll
    for (int c = 0; c < 8; ++c) {
      float4 o;
      o.x = fmaxf(r[4 * c + 0] - tau, 0.0f);
      o.y = fmaxf(r[4 * c + 1] - tau, 0.0f);
      o.z = fmaxf(r[4 * c + 2] - tau, 0.0f);
      o.w = fmaxf(r[4 * c + 3] - tau, 0.0f);
      op[c * 32 + lane] = o;  // global_store_b128, coalesced
    }

    buf ^= 1;
  }
}

extern "C" void kernel_launch(void* const* d_in, const int* in_sizes, int n_in,
                              void* d_out, int out_size, void* d_ws, size_t ws_size,
                              hipStream_t stream) {
  (void)n_in; (void)out_size; (void)d_ws; (void)ws_size;
  const float* x = (const float*)d_in[0];
  float* out = (float*)d_out;
  const long total = (long)in_sizes[0];
  const int rows = (int)(total / ROW_N);  // 65536 for the reference shape
  if (rows <= 0) return;
  int grid = (rows + WAVES_PER_BLOCK - 1) / WAVES_PER_BLOCK;
  if (grid > 2048) grid = 2048;  // persistent: ~4 rows/wave -> prefetch overlaps
  sparsemax_kernel<<<grid, BLOCK_THREADS, 0, stream>>>(x, out, rows);
}